// MCAttEGNN_19791209300071
// MI455X (gfx1250) — compile-verified
//
#include <hip/hip_runtime.h>

// ---------------------------------------------------------------------------
// MCAttEGNN for MI455X (gfx1250, wave32). fp32 end-to-end using
// V_WMMA_F32_16X16X4_F32 for all four GEMMs. One wave32 (= one 32-thread
// workgroup) owns a tile of 16 edges (M=16). Weights are pre-packed into
// per-lane fragment order so every B fragment is a single coalesced
// global_load_b64. 4 passes: alpha_raw+segmax -> exp+segsum ->
// v/agg/silu-MLP/coord -> finalize.
// ---------------------------------------------------------------------------

typedef __attribute__((ext_vector_type(2))) float v2f;
typedef __attribute__((ext_vector_type(8))) float v8f;

constexpr int HD  = 128;   // hidden
constexpr int NCC = 3;     // coord channels
constexpr int KVI = 137;   // H + NC*NC
constexpr int KP  = 140;   // KVI padded to multiple of 4
constexpr int C1N = 512;   // 4*H

constexpr int NCH_Q  = HD / 4;   // 32  K-chunks for Wq
constexpr int NCH_KV = KP / 4;   // 35  K-chunks for Wk/Wv
constexpr int NCH_C1 = HD / 4;   // 32  K-chunks for Wc1
constexpr int NCH_C2 = C1N / 4;  // 128 K-chunks for Wc2p

// ---- WMMA helpers ---------------------------------------------------------

__device__ inline v8f wmma4(v2f a, v2f b, v8f c) {
  // D = A(16x4,f32) * B(4x16,f32) + C(16x16,f32)
  return __builtin_amdgcn_wmma_f32_16x16x4_f32(
      false, a, false, b, (short)0, c, false, false);
}

// A-matrix 16x4 f32 layout: lanes 0-15 -> M=lane, K={k0,k0+1};
// lanes 16-31 -> M=lane-16, K={k0+2,k0+3}.  Source: LDS tile, row stride ld.
__device__ inline v2f ldsA(const float* t, int ld, int k0, int lane) {
  int m  = lane & 15;
  int kb = k0 + ((lane >> 4) << 1);
  v2f a;
  a[0] = t[m * ld + kb];
  a[1] = t[m * ld + kb + 1];
  return a;
}

// Packed B fragment: Bp[((c*2 + half)*ncols + ncol)*2 + {0,1}] holds the two
// K-values this lane needs -> one global_load_b64, contiguous per half-wave.
__device__ inline v2f glbBp(const float* __restrict__ Bp, int ncols, int c,
                            int ncol, int half) {
  const float* p = Bp + ((((size_t)c * 2 + half) * ncols + ncol) << 1);
  v2f b;
  b[0] = p[0];
  b[1] = p[1];
  return b;
}

__device__ inline v8f v8_bcast(float x) {
  v8f r;
  for (int i = 0; i < 8; ++i) r[i] = x;
  return r;
}

__device__ inline void atomicMaxFloat(float* addr, float val) {
  if (val >= 0.f) atomicMax((int*)addr, __float_as_int(val));
  else            atomicMin((unsigned int*)addr, __float_as_uint(val));
}

// ---- weight prep: pack all weights into WMMA-B fragment order -------------
// Also deinterleaves Wkv (k=even cols, v=odd cols), zero-pads K 137->140 and
// Wc2 N 3->16.

__global__ void kw_prep(const float* __restrict__ Wq,
                        const float* __restrict__ Wkv,
                        const float* __restrict__ bkv,
                        const float* __restrict__ Wc1,
                        const float* __restrict__ Wc2,
                        float* __restrict__ Wq_p, float* __restrict__ Wk_p,
                        float* __restrict__ Wv_p, float* __restrict__ bk,
                        float* __restrict__ bv, float* __restrict__ Wc1_p,
                        float* __restrict__ Wc2_p) {
  int i = blockIdx.x * 256 + threadIdx.x;

  if (i < NCH_Q * 2 * HD) {            // Wq: K=128, N=128
    int c = i >> 8, rem = i & 255, half = rem >> 7, n = rem & 127;
    int k = 4 * c + 2 * half;
    Wq_p[2 * i]     = Wq[(size_t)k * HD + n];
    Wq_p[2 * i + 1] = Wq[(size_t)(k + 1) * HD + n];
  }
  if (i < NCH_KV * 2 * HD) {           // Wk/Wv: K=140(pad), N=128
    int c = i >> 8, rem = i & 255, half = rem >> 7, n = rem & 127;
    int k = 4 * c + 2 * half;
    float ka = 0.f, kb = 0.f, va = 0.f, vb = 0.f;
    if (k < KVI) {
      ka = Wkv[(size_t)k * (2 * HD) + 2 * n];
      va = Wkv[(size_t)k * (2 * HD) + 2 * n + 1];
    }
    if (k + 1 < KVI) {
      kb = Wkv[(size_t)(k + 1) * (2 * HD) + 2 * n];
      vb = Wkv[(size_t)(k + 1) * (2 * HD) + 2 * n + 1];
    }
    Wk_p[2 * i] = ka; Wk_p[2 * i + 1] = kb;
    Wv_p[2 * i] = va; Wv_p[2 * i + 1] = vb;
  }
  if (i < HD) { bk[i] = bkv[2 * i]; bv[i] = bkv[2 * i + 1]; }
  if (i < NCH_C1 * 2 * C1N) {          // Wc1: K=128, N=512
    int c = i >> 10, rem = i & 1023, half = rem >> 9, n = rem & 511;
    int k = 4 * c + 2 * half;
    Wc1_p[2 * i]     = Wc1[(size_t)k * C1N + n];
    Wc1_p[2 * i + 1] = Wc1[(size_t)(k + 1) * C1N + n];
  }
  if (i < NCH_C2 * 2 * 16) {           // Wc2: K=512, N=3 padded to 16
    int c = i >> 5, rem = i & 31, half = rem >> 4, n = rem & 15;
    int k = 4 * c + 2 * half;
    float a = (n < NCC) ? Wc2[k * NCC + n] : 0.f;
    float b = (n < NCC) ? Wc2[(k + 1) * NCC + n] : 0.f;
    Wc2_p[2 * i] = a;
    Wc2_p[2 * i + 1] = b;
  }
}

// ---- init: h_out = h (atomics accumulate into it), node accumulators ------

__global__ void k_init(const float* __restrict__ h, float* __restrict__ hout,
                       float* __restrict__ nmax, float* __restrict__ nden,
                       float* __restrict__ cagg, int N) {
  long i = (long)blockIdx.x * 256 + threadIdx.x;
  long total = (long)N * HD;
  if (i < total) hout[i] = h[i];
  if (i < N) { nmax[i] = __int_as_float(0xff800000); nden[i] = 0.f; }
  if (i < (long)N * 9) cagg[i] = 0.f;
}

// ---- pass A: alpha_raw = (h[row]Wq+bq) . (feat Wk+bk); segment max --------
// One wave32 workgroup per 16-edge tile; __syncthreads degenerates to a
// waitcnt (single-wave workgroups have no barrier resource).

__global__ __launch_bounds__(32) void ka_alpha(
    const float* __restrict__ h, const float* __restrict__ coord,
    const int* __restrict__ row, const int* __restrict__ col,
    const float* __restrict__ Wq_p, const float* __restrict__ bq,
    const float* __restrict__ Wk_p, const float* __restrict__ bk,
    float* __restrict__ araw, float* __restrict__ nmax, int E) {
  __shared__ float hrow[16 * HD];
  __shared__ float feat[16 * KP];
  __shared__ int   idx[32];

  const int lane = threadIdx.x;
  const int tile = blockIdx.x;
  const int e0 = tile << 4;
  const int valid = min(16, E - e0);

  {
    int m = lane & 15;
    int e = e0 + min(m, valid - 1);
    idx[lane] = (lane < 16) ? row[e] : col[e];   // [0..15]=row, [16..31]=col
  }
  __syncthreads();

  for (int f = lane; f < 16 * (HD / 4); f += 32) {       // h[row] tile
    int m = f >> 5, kq = f & 31;
    ((float4*)hrow)[f] = ((const float4*)(h + (size_t)idx[m] * HD))[kq];
  }
  for (int f = lane; f < 16 * HD; f += 32) {             // h[col] -> feat 9..136
    int m = f >> 7, k = f & 127;
    feat[m * KP + 9 + k] = h[(size_t)idx[16 + m] * HD + k];
  }
  if (lane < 16) {                                       // radial + zero pad
    int m = lane;
    const float* cr = coord + (size_t)idx[m] * 9;
    const float* cc = coord + (size_t)idx[16 + m] * 9;
    float d[9];
    for (int i = 0; i < 9; ++i) d[i] = cr[i] - cc[i];
    for (int c = 0; c < 3; ++c)
      for (int f2 = 0; f2 < 3; ++f2)
        feat[m * KP + c * 3 + f2] =
            d[c * 3] * d[f2 * 3] + d[c * 3 + 1] * d[f2 * 3 + 1] +
            d[c * 3 + 2] * d[f2 * 3 + 2];
    feat[m * KP + 137] = 0.f; feat[m * KP + 138] = 0.f; feat[m * KP + 139] = 0.f;
  }
  __syncthreads();

  float p[8];
  for (int r = 0; r < 8; ++r) p[r] = 0.f;
  const int nloc = lane & 15;
  const int half = lane >> 4;

  for (int nt = 0; nt < 8; ++nt) {
    const int ncol = nt * 16 + nloc;
    v8f qa   = v8_bcast(bq[ncol]);
    v8f kacc = v8_bcast(bk[ncol]);
    for (int c = 0; c < NCH_Q; ++c)
      qa = wmma4(ldsA(hrow, HD, 4 * c, lane), glbBp(Wq_p, HD, c, ncol, half),
                 qa);
    for (int c = 0; c < NCH_KV; ++c)
      kacc = wmma4(ldsA(feat, KP, 4 * c, lane), glbBp(Wk_p, HD, c, ncol, half),
                   kacc);
    for (int r = 0; r < 8; ++r) p[r] += qa[r] * kacc[r];
  }
  // reduce q.k over the 16 N-lanes of each half (xor<16 stays in the half)
  for (int r = 0; r < 8; ++r) {
    float s = p[r];
    s += __shfl_xor(s, 1, 32);
    s += __shfl_xor(s, 2, 32);
    s += __shfl_xor(s, 4, 32);
    s += __shfl_xor(s, 8, 32);
    p[r] = s;
  }
  if (nloc == 0) {
    for (int r = 0; r < 8; ++r) {
      int m = r + (half << 3);
      if (m < valid) {
        araw[e0 + m] = p[r];
        atomicMaxFloat(&nmax[idx[m]], p[r]);
      }
    }
  }
}

// ---- pass B: ex = exp(alpha - max[row]); segment sum ----------------------

__global__ void kb_exp(const int* __restrict__ row,
                       const float* __restrict__ nmax,
                       float* __restrict__ araw, float* __restrict__ nden,
                       int E) {
  int e = blockIdx.x * 256 + threadIdx.x;
  if (e < E) {
    int r = row[e];
    float mx = nmax[r];
    if (!(mx > -3.0e38f && mx < 3.0e38f)) mx = 0.f;  // isfinite guard
    float ex = expf(araw[e] - mx);
    araw[e] = ex;
    atomicAdd(&nden[r], ex);
  }
}

// ---- pass C: v, h-aggregation, silu(v@Wc1)@Wc2, coord aggregation ---------

__global__ __launch_bounds__(32) void kc_agg(
    const float* __restrict__ h, const float* __restrict__ coord,
    const int* __restrict__ row, const int* __restrict__ col,
    const float* __restrict__ Wv_p, const float* __restrict__ bv,
    const float* __restrict__ Wc1_p, const float* __restrict__ Wc2_p,
    const float* __restrict__ ex, const float* __restrict__ nden,
    float* __restrict__ hout, float* __restrict__ cagg,
    float* __restrict__ alpha_out, int E) {
  __shared__ float feat[16 * KP];
  __shared__ float vbuf[16 * HD];
  __shared__ float s1[16 * 16];
  __shared__ float al[16];
  __shared__ float cd[16 * 9];
  __shared__ int   idx[32];

  const int lane = threadIdx.x;
  const int tile = blockIdx.x;
  const int e0 = tile << 4;
  const int valid = min(16, E - e0);

  {
    int m = lane & 15;
    int e = e0 + min(m, valid - 1);
    idx[lane] = (lane < 16) ? row[e] : col[e];
  }
  __syncthreads();

  for (int f = lane; f < 16 * HD; f += 32) {
    int m = f >> 7, k = f & 127;
    feat[m * KP + 9 + k] = h[(size_t)idx[16 + m] * HD + k];
  }
  if (lane < 16) {
    int m = lane;
    const float* cr = coord + (size_t)idx[m] * 9;
    const float* cc = coord + (size_t)idx[16 + m] * 9;
    float d[9];
    for (int i = 0; i < 9; ++i) { d[i] = cr[i] - cc[i]; cd[m * 9 + i] = d[i]; }
    for (int c = 0; c < 3; ++c)
      for (int f2 = 0; f2 < 3; ++f2)
        feat[m * KP + c * 3 + f2] =
            d[c * 3] * d[f2 * 3] + d[c * 3 + 1] * d[f2 * 3 + 1] +
            d[c * 3 + 2] * d[f2 * 3 + 2];
    feat[m * KP + 137] = 0.f; feat[m * KP + 138] = 0.f; feat[m * KP + 139] = 0.f;
    // softmax normalize (alpha=0 for tail-duplicate edges kills their atomics)
    int e = e0 + min(m, valid - 1);
    float a = 0.f;
    if (m < valid) {
      a = ex[e] / nden[idx[m]];
      alpha_out[e] = a;
    }
    al[m] = a;
  }
  __syncthreads();

  const int nloc = lane & 15;
  const int half = lane >> 4;

  // v = feat @ Wv + bv ; stage to LDS + aggregate alpha*v into h_out
  for (int nt = 0; nt < 8; ++nt) {
    const int ncol = nt * 16 + nloc;
    v8f va = v8_bcast(bv[ncol]);
    for (int c = 0; c < NCH_KV; ++c)
      va = wmma4(ldsA(feat, KP, 4 * c, lane), glbBp(Wv_p, HD, c, ncol, half),
                 va);
    for (int r = 0; r < 8; ++r) {
      int m = r + (half << 3);
      float val = va[r];
      vbuf[m * HD + ncol] = val;
      atomicAdd(&hout[(size_t)idx[m] * HD + ncol], al[m] * val);
    }
  }
  __syncthreads();

  // c2 = silu(v @ Wc1) @ Wc2p, accumulated over 32 K-chunks of 16
  v8f c2 = v8_bcast(0.f);
  for (int ct = 0; ct < 32; ++ct) {
    const int ncol = ct * 16 + nloc;
    v8f c1a = v8_bcast(0.f);
    for (int c = 0; c < NCH_C1; ++c)
      c1a = wmma4(ldsA(vbuf, HD, 4 * c, lane),
                  glbBp(Wc1_p, C1N, c, ncol, half), c1a);
    for (int r = 0; r < 8; ++r) {                 // silu, C-layout -> LDS
      float x = c1a[r];
      s1[(r + (half << 3)) * 16 + nloc] = x / (1.f + expf(-x));
    }
    __syncthreads();
    for (int cc = 0; cc < 4; ++cc)                // s1 as A-matrix K-chunk
      c2 = wmma4(ldsA(s1, 16, 4 * cc, lane),
                 glbBp(Wc2_p, 16, ct * 4 + cc, nloc, half), c2);
    __syncthreads();
  }

  // coord aggregation: trans = coord_diff * (alpha * c2) ; segment sum
  if (nloc < NCC) {
    for (int r = 0; r < 8; ++r) {
      int m = r + (half << 3);
      float cv = al[m] * c2[r];
      int rn = idx[m];
      for (int dd = 0; dd < 3; ++dd)
        atomicAdd(&cagg[(size_t)rn * 9 + nloc * 3 + dd],
                  cd[m * 9 + nloc * 3 + dd] * cv);
    }
  }
}

// ---- pass D: coord_out = coord + clip(cagg, -10, 10) ----------------------

__global__ void kd_coord(const float* __restrict__ coord,
                         const float* __restrict__ cagg,
                         float* __restrict__ coordout, long n9) {
  long i = (long)blockIdx.x * 256 + threadIdx.x;
  if (i < n9) {
    float c = cagg[i];
    c = fminf(10.f, fmaxf(-10.f, c));
    coordout[i] = coord[i] + c;
  }
}

// ---------------------------------------------------------------------------

extern "C" void kernel_launch(void* const* d_in, const int* in_sizes, int n_in,
                              void* d_out, int out_size, void* d_ws,
                              size_t ws_size, hipStream_t stream) {
  const float* h     = (const float*)d_in[0];
  const float* coord = (const float*)d_in[1];
  const int*   row   = (const int*)d_in[2];
  const int*   col   = (const int*)d_in[3];
  const float* Wq    = (const float*)d_in[4];
  const float* bq    = (const float*)d_in[5];
  const float* Wkv   = (const float*)d_in[6];
  const float* bkv   = (const float*)d_in[7];
  const float* Wc1   = (const float*)d_in[8];
  const float* Wc2   = (const float*)d_in[9];

  const int N = in_sizes[0] / HD;
  const int E = in_sizes[2];

  float* hout      = (float*)d_out;
  float* coordout  = hout + (size_t)N * HD;
  float* alpha_out = coordout + (size_t)N * 9;

  // workspace carve-up (16B-aligned regions)
  float* ws = (float*)d_ws;
  auto alloc = [&](size_t n) {
    float* p = ws;
    ws += (n + 3) & ~(size_t)3;
    return p;
  };
  float* araw  = alloc(E);
  float* nmax  = alloc(N);
  float* nden  = alloc(N);
  float* cagg  = alloc((size_t)N * 9);
  float* Wq_p  = alloc((size_t)NCH_Q * 2 * HD * 2);
  float* Wk_p  = alloc((size_t)NCH_KV * 2 * HD * 2);
  float* Wv_p  = alloc((size_t)NCH_KV * 2 * HD * 2);
  float* bk    = alloc(HD);
  float* bv    = alloc(HD);
  float* Wc1_p = alloc((size_t)NCH_C1 * 2 * C1N * 2);
  float* Wc2_p = alloc((size_t)NCH_C2 * 2 * 16 * 2);

  const int prep_n = NCH_C1 * 2 * C1N;  // largest packed region (32768)
  kw_prep<<<(prep_n + 255) / 256, 256, 0, stream>>>(
      Wq, Wkv, bkv, Wc1, Wc2, Wq_p, Wk_p, Wv_p, bk, bv, Wc1_p, Wc2_p);
  const long nh = (long)N * HD;
  k_init<<<(int)((nh + 255) / 256), 256, 0, stream>>>(h, hout, nmax, nden,
                                                      cagg, N);
  const int ntiles = (E + 15) / 16;
  ka_alpha<<<ntiles, 32, 0, stream>>>(h, coord, row, col, Wq_p, bq, Wk_p, bk,
                                      araw, nmax, E);
  kb_exp<<<(E + 255) / 256, 256, 0, stream>>>(row, nmax, araw, nden, E);
  kc_agg<<<ntiles, 32, 0, stream>>>(h, coord, row, col, Wv_p, bv, Wc1_p, Wc2_p,
                                    araw, nden, hout, cagg, alpha_out, E);
  kd_coord<<<(int)(((long)N * 9 + 255) / 256), 256, 0, stream>>>(
      coord, cagg, coordout, (long)N * 9);
}